// PackedExpertMLP_34832184770692
// MI455X (gfx1250) — compile-verified
//
#include <hip/hip_runtime.h>
#include <hip/hip_bf16.h>

// PackedExpertMLP for MI455X (gfx1250): HBM-bandwidth-bound f32 GEMV chain.
// ~380 MB of weights streamed exactly once (NT hint keeps the one-shot stream
// from thrashing L2; WGP-scope prefetch provides near-cache residency for the
// 4 trips that consume each 128B line). Dot products accumulate in exact f32
// on the matrix pipe via V_WMMA_F32_16X16X4_F32, software-pipelined so loads
// overlap WMMA issue.

#define D_MODEL 2816
#define D_FFN   704
#define PACK_G  16
#define NW      8            // waves per block
#define PFD     8            // prefetch distance in trips (256B/lane ahead)

typedef __attribute__((ext_vector_type(2))) float v2f;
typedef __attribute__((ext_vector_type(4))) float v4f;
typedef __attribute__((ext_vector_type(8))) float v8f;

__device__ __forceinline__ float gelu_tanh(float v) {
    // tanh-approx gelu, matches jax.nn.gelu(approximate=True)
    const float c0 = 0.7978845608028654f;   // sqrt(2/pi)
    const float c1 = 0.044715f;
    float t = tanhf(c0 * (v + c1 * v * v * v));
    return 0.5f * v * (1.0f + t);
}

// Two chained 16x16x4 f32 WMMAs consuming an 8-K block held as one float4 per
// lane. K->slot assignment is ours to choose (reduction is order-free): lane
// half `hi` carries k-offsets {4*hi..4*hi+3}; A and B agree on the mapping.
__device__ __forceinline__ v8f wmma8k(v8f c, v4f a, v4f b) {
    v2f a0 = {a.x, a.y}, a1 = {a.z, a.w};
    v2f b0 = {b.x, b.y}, b1 = {b.z, b.w};
    c = __builtin_amdgcn_wmma_f32_16x16x4_f32(false, a0, false, b0,
                                              (short)0, c, false, false);
    c = __builtin_amdgcn_wmma_f32_16x16x4_f32(false, a1, false, b1,
                                              (short)0, c, false, false);
    return c;
}

// ---------------------------------------------------------------------------
// Phase 1: h[j] = gelu(x . Wg[j,:]) * (x . Wu[j,:]),  j in [0, 16*704)
// Block = 16 rows of Wg AND Wu; 8 waves split K=2816 into 352-wide slices
// (44 trips of 8 K each); LDS reduce; lanes<16 finalize gelu(g)*u.
// ---------------------------------------------------------------------------
__global__ __launch_bounds__(256) void gateup_wmma_kernel(
    const float* __restrict__ x,
    const float* __restrict__ Wg,
    const float* __restrict__ Wu,
    float* __restrict__ h)
{
    __shared__ float xs[D_MODEL];
    __shared__ float pg[NW][16];
    __shared__ float pu[NW][16];

    const int tid = threadIdx.x;
    for (int i = tid; i < D_MODEL; i += 256) xs[i] = x[i];
    __syncthreads();

    const int wave = tid >> 5;
    const int lane = tid & 31;
    const int hi   = lane >> 4;          // lane half -> K sub-block
    const int row  = lane & 15;          // A-matrix row M
    const int j0   = blockIdx.x * 16;

    const int KS    = D_MODEL / NW;      // 352 K per wave
    const int TRIPS = KS / 8;            // 44 trips (8 K / trip)
    const int kbase = wave * KS;

    const v4f* gP = (const v4f*)(Wg + (size_t)(j0 + row) * D_MODEL + kbase + 4 * hi);
    const v4f* uP = (const v4f*)(Wu + (size_t)(j0 + row) * D_MODEL + kbase + 4 * hi);
    const v4f* bP = (const v4f*)(xs + kbase + 4 * hi);

    v8f cg = {};
    v8f cu = {};

    // software pipeline: registers for trip t loaded while trip t-1 computes
    v4f Ag = __builtin_nontemporal_load(gP);
    v4f Au = __builtin_nontemporal_load(uP);
    v4f Bv = bP[0];

    int t = 0;
    for (; t < TRIPS - PFD; ++t) {                 // main: unconditional prefetch
        v4f Ag1 = __builtin_nontemporal_load(gP + 2 * (t + 1));
        v4f Au1 = __builtin_nontemporal_load(uP + 2 * (t + 1));
        v4f Bv1 = bP[2 * (t + 1)];
        __builtin_prefetch(gP + 2 * (t + PFD), 0, 3);
        __builtin_prefetch(uP + 2 * (t + PFD), 0, 3);
        cg = wmma8k(cg, Ag, Bv);
        cu = wmma8k(cu, Au, Bv);
        Ag = Ag1; Au = Au1; Bv = Bv1;
    }
    for (; t < TRIPS - 1; ++t) {                   // tail: no prefetch
        v4f Ag1 = __builtin_nontemporal_load(gP + 2 * (t + 1));
        v4f Au1 = __builtin_nontemporal_load(uP + 2 * (t + 1));
        v4f Bv1 = bP[2 * (t + 1)];
        cg = wmma8k(cg, Ag, Bv);
        cu = wmma8k(cu, Au, Bv);
        Ag = Ag1; Au = Au1; Bv = Bv1;
    }
    cg = wmma8k(cg, Ag, Bv);
    cu = wmma8k(cu, Au, Bv);

    // D layout: VGPR r = row (r + 8*hi); column 0 lives in lanes 0 and 16.
    union { v8f v; float f[8]; } ag, au;
    ag.v = cg; au.v = cu;
    if (row == 0) {
#pragma unroll
        for (int r = 0; r < 8; ++r) {
            pg[wave][hi * 8 + r] = ag.f[r];
            pu[wave][hi * 8 + r] = au.f[r];
        }
    }
    __syncthreads();
    if (tid < 16) {
        float dg = 0.f, du = 0.f;
#pragma unroll
        for (int w = 0; w < NW; ++w) { dg += pg[w][tid]; du += pu[w][tid]; }
        h[j0 + tid] = gelu_tanh(dg) * du;
    }
}

// ---------------------------------------------------------------------------
// Phase 2: out[g,d] = sum_f h[g,f] * Wd[g,d,f]   (16 experts, d=2816, f=704)
// Block = one expert's 16-row group; 8 waves split f into 88-wide slices
// (11 trips of 8 f each).
// ---------------------------------------------------------------------------
__global__ __launch_bounds__(256) void down_wmma_kernel(
    const float* __restrict__ h,
    const float* __restrict__ Wd,
    float* __restrict__ out)
{
    __shared__ float hs[D_FFN];
    __shared__ float pc[NW][16];

    const int tid    = threadIdx.x;
    const int groups = D_MODEL / 16;            // 176 row-groups per expert
    const int gidx   = blockIdx.x / groups;     // expert index
    const int d0     = (blockIdx.x % groups) * 16;

    for (int i = tid; i < D_FFN; i += 256) hs[i] = h[gidx * D_FFN + i];
    __syncthreads();

    const int wave = tid >> 5;
    const int lane = tid & 31;
    const int hi   = lane >> 4;
    const int row  = lane & 15;

    const int FS    = D_FFN / NW;               // 88 f per wave
    const int TRIPS = FS / 8;                   // 11 trips
    const int fbase = wave * FS;

    const v4f* wP = (const v4f*)(Wd + ((size_t)gidx * D_MODEL + d0 + row) * D_FFN
                                    + fbase + 4 * hi);
    const v4f* bP = (const v4f*)(hs + fbase + 4 * hi);

    v8f c = {};
    v4f Aw = __builtin_nontemporal_load(wP);
    v4f Bv = bP[0];

    int t = 0;
    for (; t < TRIPS - PFD; ++t) {
        v4f Aw1 = __builtin_nontemporal_load(wP + 2 * (t + 1));
        v4f Bv1 = bP[2 * (t + 1)];
        __builtin_prefetch(wP + 2 * (t + PFD), 0, 3);
        c = wmma8k(c, Aw, Bv);
        Aw = Aw1; Bv = Bv1;
    }
    for (; t < TRIPS - 1; ++t) {
        v4f Aw1 = __builtin_nontemporal_load(wP + 2 * (t + 1));
        v4f Bv1 = bP[2 * (t + 1)];
        c = wmma8k(c, Aw, Bv);
        Aw = Aw1; Bv = Bv1;
    }
    c = wmma8k(c, Aw, Bv);

    union { v8f v; float f[8]; } ac;
    ac.v = c;
    if (row == 0) {
#pragma unroll
        for (int r = 0; r < 8; ++r) pc[wave][hi * 8 + r] = ac.f[r];
    }
    __syncthreads();
    if (tid < 16) {
        float d = 0.f;
#pragma unroll
        for (int w = 0; w < NW; ++w) d += pc[w][tid];
        out[gidx * D_MODEL + d0 + tid] = d;
    }
}

extern "C" void kernel_launch(void* const* d_in, const int* in_sizes, int n_in,
                              void* d_out, int out_size, void* d_ws, size_t ws_size,
                              hipStream_t stream) {
    const float* x  = (const float*)d_in[0];   // (1,1,2816)
    const float* Wg = (const float*)d_in[1];   // (11264, 2816)
    const float* Wu = (const float*)d_in[2];   // (11264, 2816)
    const float* Wd = (const float*)d_in[3];   // (45056, 704)
    float* out = (float*)d_out;                // (16, 2816) flat
    float* h   = (float*)d_ws;                 // 16*704 floats scratch

    gateup_wmma_kernel<<<(PACK_G * D_FFN) / 16, 256, 0, stream>>>(x, Wg, Wu, h);
    down_wmma_kernel<<<PACK_G * (D_MODEL / 16), 256, 0, stream>>>(h, Wd, out);
}